// INSIDE_7026566496362
// MI455X (gfx1250) — compile-verified
//
#include <hip/hip_runtime.h>
#include <math.h>

typedef __attribute__((ext_vector_type(2))) float v2f;
typedef __attribute__((ext_vector_type(8))) float v8f;

#define GEPS 1e-7f

// ---------------------------------------------------------------------------
// WMMA f32 16x16x4 tile GEMM: D(16x16) = A(16xK) @ B(K x *) tile at col n0.
// A row-major [16 x lda], B row-major [K x ldb]. K must be a multiple of 4.
// Per ISA 7.12.2 (wave32):
//   A frag: lane L -> M = L&15, K = (L>>4)*2 + vgpr
//   B frag: lane L -> N = L&15, K = (L>>4)*2 + vgpr
//   C/D   : lane L -> N = L&15, vgpr v -> M = v + (L>>4)*8
// ---------------------------------------------------------------------------
__device__ inline v8f wmma_tile_f32(const float* __restrict__ A, int lda,
                                    const float* __restrict__ B, int ldb,
                                    int n0, int K, int lane) {
    const int m    = lane & 15;
    const int n    = n0 + (lane & 15);
    const int koff = (lane >> 4) * 2;
    v8f c = {0.f, 0.f, 0.f, 0.f, 0.f, 0.f, 0.f, 0.f};
    for (int kk = 0; kk < K; kk += 4) {
        v2f a, b;
        a.x = A[m * lda + kk + koff];
        a.y = A[m * lda + kk + koff + 1];
        b.x = B[(kk + koff) * ldb + n];
        b.y = B[(kk + koff + 1) * ldb + n];
        c = __builtin_amdgcn_wmma_f32_16x16x4_f32(false, a, false, b,
                                                  (short)0, c, false, false);
    }
    return c;
}

__device__ inline void store_tile(float* __restrict__ D, int ldd, int n0,
                                  v8f c, const float* __restrict__ bias,
                                  bool do_tanh, int lane) {
    const int n     = n0 + (lane & 15);
    const int mbase = (lane >> 4) * 8;
    const float bv  = bias[n];
#pragma unroll
    for (int v = 0; v < 8; ++v) {
        float val = c[v] + bv;
        if (do_tanh) val = tanhf(val);
        D[(mbase + v) * ldd + n] = val;
    }
}

// ---------------------------------------------------------------------------
// Kernel 1: whole MLP in one workgroup (4 waves, 128 threads), all matmuls
// through V_WMMA_F32_16X16X4_F32. Writes p (16x512, tanh on cols>=256) and
// s (16x256, tanh) to workspace.
// ---------------------------------------------------------------------------
__global__ void __launch_bounds__(128)
mlp_wmma_kernel(const float* __restrict__ z,
                const float* __restrict__ W1, const float* __restrict__ b1,
                const float* __restrict__ W2, const float* __restrict__ b2,
                const float* __restrict__ W3, const float* __restrict__ b3,
                const float* __restrict__ W4, const float* __restrict__ b4,
                float* __restrict__ p_out, float* __restrict__ s_out) {
    __shared__ float h1[16 * 64];
    __shared__ float h2[16 * 64];
    const int lane = threadIdx.x & 31;
    const int wave = threadIdx.x >> 5;  // 0..3

    // h1 = tanh(z @ W1 + b1)     (16x64): 4 tiles, one per wave
    {
        v8f c = wmma_tile_f32(z, 64, W1, 64, wave * 16, 64, lane);
        store_tile(h1, 64, wave * 16, c, b1, true, lane);
    }
    __syncthreads();
    // h2 = tanh(h1 @ W2 + b2)    (16x64)
    {
        v8f c = wmma_tile_f32(h1, 64, W2, 64, wave * 16, 64, lane);
        store_tile(h2, 64, wave * 16, c, b2, true, lane);
    }
    __syncthreads();
    // p = h2 @ W3 + b3           (16x512): 32 tiles, 8 per wave.
    // tanh only on columns >= 256 (mu_a / mu_b blocks) -> tile index >= 16.
    for (int t = wave; t < 32; t += 4) {
        v8f c = wmma_tile_f32(h2, 64, W3, 512, t * 16, 64, lane);
        store_tile(p_out, 512, t * 16, c, b3, (t >= 16), lane);
    }
    // s = tanh(h2 @ W4 + b4)     (16x256): 16 tiles, 4 per wave
    for (int t = wave; t < 16; t += 4) {
        v8f c = wmma_tile_f32(h2, 64, W4, 256, t * 16, 64, lane);
        store_tile(s_out, 256, t * 16, c, b4, true, lane);
    }
}

// ---------------------------------------------------------------------------
// Kernel 2: bandwidth-bound streaming pass. One block per (b, h) row-plane
// (128x128 elements). Per-channel params + the a_x row are staged in LDS;
// a_y is recomputed inline (one v_exp_f32 per element) so no mask arrays
// ever touch HBM. Traffic: read x once, write a and y once -> ~402 MB total.
// ---------------------------------------------------------------------------
__global__ void __launch_bounds__(256)
mask_apply_kernel(const float* __restrict__ x,
                  const float* __restrict__ p,   // ws: 16x512
                  const float* __restrict__ s,   // ws: 16x256
                  float* __restrict__ outA,
                  float* __restrict__ outY) {
    const int bh = blockIdx.x;   // 0..2047
    const int b  = bh >> 7;
    const int h  = bh & 127;

    __shared__ float s_ax[128], s_scale[128], s_shift[128];
    __shared__ float s_mub[128], s_sgb[128];

    const int tid = threadIdx.x;
    if (tid < 128) {
        const int c = tid;
        const float scale = p[b * 512 + c];
        const float shift = p[b * 512 + 128 + c];
        const float mu_a  = p[b * 512 + 256 + c];   // already tanh'ed
        const float mu_b  = p[b * 512 + 384 + c];   // already tanh'ed
        const float sg_a  = s[b * 256 + c];
        const float sg_b  = s[b * 256 + 128 + c];
        const float mua = mu_a * 64.f + 63.5f;      // dim/2, dim/2 - 0.5
        const float sga = sg_a * 3.5f + GEPS;
        const float d   = ((float)h - mua) / sga;
        s_ax[c]    = __expf(-0.5f * d * d);
        s_scale[c] = scale;
        s_shift[c] = shift;
        s_mub[c]   = mu_b * 64.f + 63.5f;
        s_sgb[c]   = sg_b * 3.5f + GEPS;
    }
    __syncthreads();

    const size_t base = (size_t)bh * (128 * 128);
    const float4* __restrict__ x4p = (const float4*)(x + base);
    float4* __restrict__ a4p = (float4*)(outA + base);
    float4* __restrict__ y4p = (float4*)(outY + base);

    // 128 (w) * 32 (c/4) = 4096 float4 units; 256 threads x 16 iters
#pragma unroll 4
    for (int it = 0; it < 16; ++it) {
        const int u = it * 256 + tid;
        const int w = u >> 5;
        const int c = (u & 31) * 4;

        float xr[4];
        *(float4*)xr = x4p[u];
        float ar[4], yr[4];
        const float wf = (float)w;
#pragma unroll
        for (int i = 0; i < 4; ++i) {
            const float dd = (wf - s_mub[c + i]) / s_sgb[c + i];
            const float ay = __expf(-0.5f * dd * dd);
            const float a  = s_ax[c + i] * ay;
            ar[i] = a;
            yr[i] = s_scale[c + i] * (xr[i] * a) + s_shift[c + i];
        }
        a4p[u] = *(float4*)ar;
        y4p[u] = *(float4*)yr;
    }
}

// ---------------------------------------------------------------------------
// Host launcher. Workspace layout (floats): p[16*512] then s[16*256] = 48 KB.
// d_out = [ a (33554432 floats) | y (33554432 floats) ].
// ---------------------------------------------------------------------------
extern "C" void kernel_launch(void* const* d_in, const int* in_sizes, int n_in,
                              void* d_out, int out_size, void* d_ws, size_t ws_size,
                              hipStream_t stream) {
    const float* x  = (const float*)d_in[0];
    const float* z  = (const float*)d_in[1];
    const float* W1 = (const float*)d_in[2];
    const float* b1 = (const float*)d_in[3];
    const float* W2 = (const float*)d_in[4];
    const float* b2 = (const float*)d_in[5];
    const float* W3 = (const float*)d_in[6];
    const float* b3 = (const float*)d_in[7];
    const float* W4 = (const float*)d_in[8];
    const float* b4 = (const float*)d_in[9];

    float* ws_f = (float*)d_ws;
    float* p    = ws_f;               // 16*512
    float* s    = ws_f + 16 * 512;    // 16*256

    float* outA = (float*)d_out;
    float* outY = outA + (size_t)out_size / 2;

    mlp_wmma_kernel<<<1, 128, 0, stream>>>(z, W1, b1, W2, b2, W3, b3, W4, b4, p, s);
    mask_apply_kernel<<<2048, 256, 0, stream>>>(x, p, s, outA, outY);
}